// DetectionLoss_50500225467004
// MI455X (gfx1250) — compile-verified
//
#include <hip/hip_runtime.h>
#include <math.h>

#define N_CLASSES 80
#define N_BINS    16
#define B_        32
#define T_        64
#define C_TOT     144          // 4*N_BINS + N_CLASSES
#define NT_FOCAL  (3 * B_ * T_)   // 6144 targets (layer, b, t)
#define NT_DFL    (3 * B_)        // 96 items (layer, b)

typedef __attribute__((ext_vector_type(2))) float v2f;
typedef __attribute__((ext_vector_type(8))) float v8f;

__device__ __forceinline__ float wave_max(float v) {
#pragma unroll
    for (int m = 16; m >= 1; m >>= 1) v = fmaxf(v, __shfl_xor(v, m, 32));
    return v;
}
__device__ __forceinline__ float wave_sum(float v) {
#pragma unroll
    for (int m = 16; m >= 1; m >>= 1) v += __shfl_xor(v, m, 32);
    return v;
}

// ---------------------------------------------------------------------------
// Stage 1a: focal classification loss, one wave32 per (layer, b, t) target.
// Lane L gathers class logits L, L+32, L+64 (channel-strided); shfl reductions
// give the stable 80-way log-softmax. Result per target -> ws_focal[wid].
// ---------------------------------------------------------------------------
__global__ __launch_bounds__(256)
void focal_kernel(const float* __restrict__ f0, const float* __restrict__ f1,
                  const float* __restrict__ f2,
                  const float* __restrict__ tgt_box,
                  const int* __restrict__ tgt_cls,
                  const int* __restrict__ tgt_layer,
                  float* __restrict__ ws_focal) {
    const int wid  = blockIdx.x * 8 + (threadIdx.x >> 5);
    const int lane = threadIdx.x & 31;
    if (wid >= NT_FOCAL) return;

    const int layer = wid / (B_ * T_);
    const int rem   = wid - layer * (B_ * T_);
    const int b     = rem / T_;
    const int t     = rem - b * T_;

    const float* feat; int S;
    if (layer == 0)      { feat = f0; S = 80; }
    else if (layer == 1) { feat = f1; S = 40; }
    else                 { feat = f2; S = 20; }
    const int S2 = S * S;

    const int bt = b * T_ + t;
    const float cx = tgt_box[bt * 4 + 0];
    const float cy = tgt_box[bt * 4 + 1];
    int fx = (int)(cx * (float)S); fx = max(0, min(fx, S - 1));
    int fy = (int)(cy * (float)S); fy = max(0, min(fy, S - 1));

    // class logits start at channel 4*N_BINS = 64
    const float* base = feat + ((size_t)(b * C_TOT + 4 * N_BINS) * S + fy) * S + fx;

    const float l0 = base[(size_t)(lane)      * S2];
    const float l1 = base[(size_t)(lane + 32) * S2];
    const float l2 = (lane < 16) ? base[(size_t)(lane + 64) * S2] : -INFINITY;

    const float m = wave_max(fmaxf(fmaxf(l0, l1), l2));
    float e = expf(l0 - m) + expf(l1 - m) + ((lane < 16) ? expf(l2 - m) : 0.0f);
    e = wave_sum(e);
    const float logZ = m + logf(e);

    const int c    = tgt_cls[bt];           // wave-uniform
    const int slot = c >> 5;                // which of l0/l1/l2 holds class c
    const float pick = (slot == 0) ? l0 : ((slot == 1) ? l1 : l2);
    const float lt   = __shfl(pick, c & 31, 32);

    const float ce    = logZ - lt;
    const float pt    = expf(-ce);
    const float omp   = 1.0f - pt;
    const float focal = omp * omp * ce;

    if (lane == 0)
        ws_focal[wid] = (tgt_layer[bt] == layer) ? focal : 0.0f;
}

// ---------------------------------------------------------------------------
// Stage 1b: DFL box loss. Faithful to the reference's indentation bug: only
// the LAST target per image matching this layer contributes. 96 threads total.
// ---------------------------------------------------------------------------
__global__ __launch_bounds__(96)
void dfl_kernel(const float* __restrict__ f0, const float* __restrict__ f1,
                const float* __restrict__ f2,
                const float* __restrict__ tgt_box,
                const int* __restrict__ tgt_layer,
                float* __restrict__ ws_dfl) {
    const int idx = blockIdx.x * blockDim.x + threadIdx.x;
    if (idx >= NT_DFL) return;
    const int layer = idx / B_;
    const int b     = idx - layer * B_;

    const float* feat; int S;
    if (layer == 0)      { feat = f0; S = 80; }
    else if (layer == 1) { feat = f1; S = 40; }
    else                 { feat = f2; S = 20; }
    const int S2 = S * S;

    int last = -1;
    for (int t = 0; t < T_; ++t)
        if (tgt_layer[b * T_ + t] == layer) last = t;
    if (last < 0) { ws_dfl[idx] = 0.0f; return; }

    const int bt = b * T_ + last;
    const float cx = tgt_box[bt * 4 + 0];
    const float cy = tgt_box[bt * 4 + 1];
    const float w  = tgt_box[bt * 4 + 2];
    const float h  = tgt_box[bt * 4 + 3];
    int fx = (int)(cx * (float)S); fx = max(0, min(fx, S - 1));
    int fy = (int)(cy * (float)S); fy = max(0, min(fy, S - 1));

    const float lw = fmaxf(w, 0.0f) * 0.5f;
    const float lh = fmaxf(h, 0.0f) * 0.5f;
    const float gt[4] = { lw * (float)S, lh * (float)S, lw * (float)S, lh * (float)S };

    const float* base = feat + ((size_t)(b * C_TOT) * S + fy) * S + fx;

    float acc = 0.0f;
#pragma unroll
    for (int r = 0; r < 4; ++r) {
        float logit[N_BINS];
        float m = -INFINITY;
#pragma unroll
        for (int k = 0; k < N_BINS; ++k) {
            logit[k] = base[(size_t)(r * N_BINS + k) * S2];
            m = fmaxf(m, logit[k]);
        }
        float s = 0.0f;
#pragma unroll
        for (int k = 0; k < N_BINS; ++k) s += expf(logit[k] - m);
        const float logZ = m + logf(s);

        float tt = fminf(fmaxf(gt[r], 0.0f), (float)(N_BINS - 1) - 1e-6f);
        const float lo = floorf(tt);
        const float wl = lo + 1.0f - tt;
        const float wr = tt - lo;
        const int lo_i = (int)lo;
        const int hi_i = min(lo_i + 1, N_BINS - 1);
        const float lps_lo = logit[lo_i] - logZ;
        const float lps_hi = logit[hi_i] - logZ;
        acc += -(lps_lo * wl + lps_hi * wr);
    }
    ws_dfl[idx] = acc;
}

// ---------------------------------------------------------------------------
// Stage 2: deterministic final reduction on ONE wave32 using the f32 WMMA as
// an accumulate engine:  D = A x Ones(4x16) + C.  Every D column equals the
// per-row running sum, so sum(D) == 16 * sum(all inputs), in full fp32.
// 6144 focal partials = 96 iterations x 64 values (one v2f per lane).
// ---------------------------------------------------------------------------
__global__ __launch_bounds__(32)
void reduce_kernel(const float* __restrict__ ws_focal,
                   const float* __restrict__ ws_dfl,
                   float* __restrict__ out) {
    const int lane = threadIdx.x;

    v2f ones; ones[0] = 1.0f; ones[1] = 1.0f;
    v8f acc = {};
#pragma unroll 4
    for (int it = 0; it < NT_FOCAL / 64; ++it) {
        const v2f a = *(const v2f*)(ws_focal + it * 64 + lane * 2);
        // (neg_a, A, neg_b, B, c_mod, C, reuse_a, reuse_b)
        acc = __builtin_amdgcn_wmma_f32_16x16x4_f32(
                  false, a, false, ones, (short)0, acc, false, false);
    }
    float s = 0.0f;
#pragma unroll
    for (int i = 0; i < 8; ++i) s += acc[i];
    s = wave_sum(s);
    const float cls_tot = s * (1.0f / 16.0f);

    float bx = ws_dfl[lane] + ws_dfl[lane + 32] + ws_dfl[lane + 64];
    bx = wave_sum(bx);

    if (lane == 0) {
        out[0] = cls_tot + bx;  // total (CLS_W = BOX_W = 1)
        out[1] = cls_tot;
        out[2] = bx;
    }
}

extern "C" void kernel_launch(void* const* d_in, const int* in_sizes, int n_in,
                              void* d_out, int out_size, void* d_ws, size_t ws_size,
                              hipStream_t stream) {
    const float* f0        = (const float*)d_in[0];
    const float* f1        = (const float*)d_in[1];
    const float* f2        = (const float*)d_in[2];
    const float* tgt_box   = (const float*)d_in[3];
    const int*   tgt_cls   = (const int*)d_in[4];
    const int*   tgt_layer = (const int*)d_in[5];
    float* out = (float*)d_out;

    float* ws_focal = (float*)d_ws;          // 6144 floats
    float* ws_dfl   = ws_focal + NT_FOCAL;   // 96 floats

    // 6144 waves, 8 waves (256 threads) per block -> 768 blocks
    focal_kernel<<<NT_FOCAL / 8, 256, 0, stream>>>(f0, f1, f2, tgt_box, tgt_cls,
                                                   tgt_layer, ws_focal);
    dfl_kernel<<<1, 96, 0, stream>>>(f0, f1, f2, tgt_box, tgt_layer, ws_dfl);
    reduce_kernel<<<1, 32, 0, stream>>>(ws_focal, ws_dfl, out);
}